// Covariance_62182536512305
// MI455X (gfx1250) — compile-verified
//
#include <hip/hip_runtime.h>
#include <hip/hip_bf16.h>
#include <stdint.h>

typedef __attribute__((ext_vector_type(2))) float v2f;
typedef __attribute__((ext_vector_type(4))) float v4f;
typedef __attribute__((ext_vector_type(8))) float v8f;

#define B_DIM   128
#define L_DIM   512
#define D_DIM   128
#define STRIDE  8
#define W_DIM   (L_DIM / STRIDE)            /* 64 windows per batch */
#define TRI     ((D_DIM * (D_DIM - 1)) / 2) /* 8128 */
#define WAVES_PER_BLOCK 8

__global__ __launch_bounds__(256)
void Covariance_62182536512305_kernel(const float* __restrict__ feature,
                                      float* __restrict__ out) {
    // One wave32 handles one (b, w) window: an 8x128 contiguous slab.
    __shared__ float lds[WAVES_PER_BLOCK * STRIDE * D_DIM]; // 8 * 8*128 f32 = 32 KB

    const int lane = threadIdx.x & 31;
    const int wid  = threadIdx.x >> 5;
    const int half = lane >> 4;   // 0: lanes 0-15, 1: lanes 16-31
    const int lcol = lane & 15;

    // Window index is wave-uniform; readfirstlane makes that provable so the
    // store base stays in SGPRs and stores use saddr + zext(u32 voffset).
    const int tile = __builtin_amdgcn_readfirstlane(
        (int)(blockIdx.x * WAVES_PER_BLOCK + wid));        // 0..8191
    const int b = tile >> 6;               // / W_DIM
    const int w = tile & (W_DIM - 1);

    float* xw = lds + wid * (STRIDE * D_DIM);
    const float* src = feature + ((size_t)b * L_DIM + (size_t)w * STRIDE) * D_DIM;
    char* const ob   = (char*)(out + (size_t)tile * TRI);  // uniform output base

    // ---- Phase 1: load 8x128 f32 (contiguous 4 KB) into LDS via 128-bit NT loads ----
    #pragma unroll
    for (int r = 0; r < 8; ++r) {
        const int e4 = r * 32 + lane;               // 16-byte chunk index 0..255
        ((v4f*)xw)[e4] = __builtin_nontemporal_load(&((const v4f*)src)[e4]);
    }
    __syncthreads();

    // ---- Phase 2: center columns; each lane owns 4 columns ----
    #pragma unroll
    for (int cc = 0; cc < 4; ++cc) {
        const int c = lane * 4 + cc;
        float sum = 0.0f;
        #pragma unroll
        for (int s = 0; s < STRIDE; ++s) sum += xw[s * D_DIM + c];
        const float mean = sum * (1.0f / STRIDE);
        #pragma unroll
        for (int s = 0; s < STRIDE; ++s) xw[s * D_DIM + c] -= mean;
    }
    __syncthreads();

    // ---- Phase 3: cov/7 = (xc^T/7) * xc via V_WMMA_F32_16X16X4_F32 (K=8 = 2 ops) ----
    const float invdf = 1.0f / (float)(STRIDE - 1);
    const int   s0    = half * 2;  // A/B 32-bit layout: lanes16-31 carry K=2,3

    for (int ti = 0; ti < 8; ++ti) {
        const int colA = ti * 16 + lcol;
        v2f ua0, ua1;                      // unscaled (used as B on diagonal tile)
        ua0.x = xw[(s0    ) * D_DIM + colA];
        ua0.y = xw[(s0 + 1) * D_DIM + colA];
        ua1.x = xw[(s0 + 4) * D_DIM + colA];
        ua1.y = xw[(s0 + 5) * D_DIM + colA];
        const v2f a0 = ua0 * invdf;        // fold /7 into A once per ti
        const v2f a1 = ua1 * invdf;

        // Triangular row byte-offsets ig*(ig-1)/2*4 for this ti (ti/half only);
        // incremental: one add per row.
        const unsigned ig0 = (unsigned)(ti * 16 + half * 8);
        unsigned rowB[8];
        rowB[0] = ((ig0 * (ig0 - 1)) >> 1) * 4u;
        #pragma unroll
        for (int v = 1; v < 8; ++v) rowB[v] = rowB[v - 1] + (ig0 + v - 1) * 4u;

        // --- Diagonal tile (tj == ti): B == unscaled A, predicated stores ---
        {
            v8f c = {0.f, 0.f, 0.f, 0.f, 0.f, 0.f, 0.f, 0.f};
            c = __builtin_amdgcn_wmma_f32_16x16x4_f32(
                    false, a0, false, ua0, (short)0, c, false, false);
            c = __builtin_amdgcn_wmma_f32_16x16x4_f32(
                    false, a1, false, ua1, (short)0, c, false, false);
            const unsigned jg  = (unsigned)(ti * 16 + lcol);
            const unsigned jgB = jg * 4u;
            #pragma unroll
            for (int v = 0; v < 8; ++v) {
                if (ig0 + (unsigned)v > jg) {
                    __builtin_nontemporal_store(c[v], (float*)(ob + (rowB[v] + jgB)));
                }
            }
        }

        // --- Off-diagonal tiles (tj < ti): every element is strictly lower ---
        for (int tj = 0; tj < ti; ++tj) {
            const int colB = tj * 16 + lcol;
            v2f b0, b1;
            b0.x = xw[(s0    ) * D_DIM + colB];
            b0.y = xw[(s0 + 1) * D_DIM + colB];
            b1.x = xw[(s0 + 4) * D_DIM + colB];
            b1.y = xw[(s0 + 5) * D_DIM + colB];

            v8f c = {0.f, 0.f, 0.f, 0.f, 0.f, 0.f, 0.f, 0.f};
            c = __builtin_amdgcn_wmma_f32_16x16x4_f32(
                    false, a0, false, b0, (short)0, c, false, false);
            c = __builtin_amdgcn_wmma_f32_16x16x4_f32(
                    false, a1, false, b1, (short)0, c, false, false);

            const unsigned jgB = (unsigned)(tj * 16 + lcol) * 4u;
            #pragma unroll
            for (int v = 0; v < 8; ++v) {   // ig >= 16*ti > jg always here
                __builtin_nontemporal_store(c[v], (float*)(ob + (rowB[v] + jgB)));
            }
        }
    }
}

extern "C" void kernel_launch(void* const* d_in, const int* in_sizes, int n_in,
                              void* d_out, int out_size, void* d_ws, size_t ws_size,
                              hipStream_t stream) {
    const float* feature = (const float*)d_in[0];
    float* out = (float*)d_out;
    // 8192 windows / 8 waves per block = 1024 blocks of 256 threads (8 wave32s)
    dim3 grid((B_DIM * W_DIM) / WAVES_PER_BLOCK);
    dim3 block(WAVES_PER_BLOCK * 32);
    Covariance_62182536512305_kernel<<<grid, block, 0, stream>>>(feature, out);
}